// InvariantAttnPool_14714557956428
// MI455X (gfx1250) — compile-verified
//
#include <hip/hip_runtime.h>

typedef __attribute__((ext_vector_type(2))) float v2f;
typedef __attribute__((ext_vector_type(4))) float f4;
typedef __attribute__((ext_vector_type(8))) float v8f;
typedef __attribute__((ext_vector_type(4))) unsigned int v4u;
typedef __attribute__((ext_vector_type(4))) int v4i;
typedef __attribute__((ext_vector_type(8))) int v8i;

constexpr int Bc = 16;
constexpr int Cc = 256;
constexpr int Lc = 8192;
constexpr int Dc = 512;
constexpr int Ac = 64;
constexpr int TL = 128;   // l-tile per block

#define IAP_HAS_TDM __has_builtin(__builtin_amdgcn_tensor_load_to_lds)

// ---------------------------------------------------------------------------
// Kernel 1: scale_qk = <wq,wk>/sqrt(64) -> ws[0];  u = w_out @ wv -> ws[16..527]
// u computed with V_WMMA_F32_16X16X4_F32 (f32 in/out, exact precision).
// ---------------------------------------------------------------------------
__global__ __launch_bounds__(256) void iap_prep_kernel(
    const float* __restrict__ wq, const float* __restrict__ wk,
    const float* __restrict__ wv, const float* __restrict__ w_out,
    float* __restrict__ ws)
{
    __shared__ float red[Ac];
    const int tid = threadIdx.x;

    if (tid < Ac) red[tid] = wq[tid] * wk[tid];
    __syncthreads();
    if (tid == 0) {
        float s = 0.f;
        for (int i = 0; i < Ac; ++i) s += red[i];
        ws[0] = s * 0.125f;                    // <wq,wk> / sqrt(64)
    }

    const int wave = tid >> 5;
    const int lane = tid & 31;
#if __has_builtin(__builtin_amdgcn_wmma_f32_16x16x4_f32)
    // 32 row-tiles of 16; 8 waves -> 4 tiles each. K=64 in 16 chunks of 4.
    // A (16x4 f32): lanes 0-15 hold M=lane K={0,1}; lanes 16-31 K={2,3}.
    // B (4x16 f32): broadcast wv over all N columns (lane halves split K).
    for (int t = wave; t < Dc / 16; t += 8) {
        const int d0    = t * 16;
        const int row   = d0 + (lane & 15);
        const int khalf = (lane >> 4) << 1;    // 0 or 2
        v8f acc = {};
        for (int kc = 0; kc < Ac / 4; ++kc) {
            const int k0 = kc * 4 + khalf;
            v2f a; a.x = w_out[row * Ac + k0]; a.y = w_out[row * Ac + k0 + 1];
            v2f b; b.x = wv[k0];               b.y = wv[k0 + 1];
            acc = __builtin_amdgcn_wmma_f32_16x16x4_f32(
                false, a, false, b, (short)0, acc, false, false);
        }
        // D layout: VGPR r, lanes 0-15 -> M=r; lanes 16-31 -> M=8+r.
        if ((lane & 15) == 0) {
            const int mb = d0 + ((lane >> 4) << 3);
            #pragma unroll
            for (int r = 0; r < 8; ++r) ws[16 + mb + r] = acc[r];
        }
    }
#else
    for (int d = tid; d < Dc; d += 256) {
        float s = 0.f;
        for (int a = 0; a < Ac; ++a) s += w_out[d * Ac + a] * wv[a];
        ws[16 + d] = s;
    }
#endif
}

// ---------------------------------------------------------------------------
// Kernel 2: per (b, l-tile): TDM-stage h_v[b, :, l0:l0+TL] into LDS (one
// tensor_load_to_lds per block, single HBM read), softmax-pool over C per
// column, write psi = pooled * u (rank-1) with non-temporal float4 stores.
// ---------------------------------------------------------------------------
__global__ __launch_bounds__(256) void iap_pool_kernel(
    const float* __restrict__ hv, const float* __restrict__ ws,
    float* __restrict__ out)
{
    __shared__ __align__(16) float tile[Cc][TL];   // 128 KB (<< 320 KB/WGP)
    __shared__ __align__(16) float u_sh[Dc];
    __shared__ __align__(16) float pooled_sh[TL];

    const int tid = threadIdx.x;
    const int b   = blockIdx.x / (Lc / TL);
    const int l0  = (blockIdx.x % (Lc / TL)) * TL;

    const float scale_qk = ws[0];
    const size_t hv_base = (size_t)b * Cc * Lc + l0;

#if IAP_HAS_TDM
    // Wave 0 issues one Tensor-Data-Mover DMA for the whole 128 KB tile:
    // 2D tile, 256 rows x 512 B, row stride 8192 elements, into LDS.
    if (tid < 32) {
        const unsigned long long ga =
            (unsigned long long)(const void*)(hv + hv_base);
        const unsigned int lds_base =
            (unsigned int)(unsigned long long)(const void*)&tile[0][0];
        v4u g0;
        g0.x = 1u;                                       // count=1 (valid D#)
        g0.y = lds_base;                                 // lds_addr (bytes)
        g0.z = (unsigned int)ga;                         // global_addr[31:0]
        g0.w = (unsigned int)(ga >> 32) | (2u << 30);    // addr[56:32] | type=2
        v8i g1;
        g1.s0 = (int)(2u << 16);                         // data_size = 4 bytes
        g1.s1 = (int)(((unsigned)TL) << 16);             // tensor_dim0 = 128
        g1.s2 = (int)(((unsigned)Cc) << 16);             // tensor_dim1 = 256
        g1.s3 = (int)(((unsigned)TL) << 16);             // tile_dim0   = 128
        g1.s4 = (int)Cc;                                 // tile_dim1   = 256
        g1.s5 = (int)Lc;                                 // dim0_stride = 8192
        g1.s6 = 0;
        g1.s7 = 0;
        v4i z4 = {};                                     // groups 2/3: <=2D
#if defined(__clang_major__) && (__clang_major__ >= 23)
        v8i z8 = {};
        __builtin_amdgcn_tensor_load_to_lds(g0, g1, z4, z4, z8, 0);
#else
        __builtin_amdgcn_tensor_load_to_lds(g0, g1, z4, z4, 0);
#endif
        __builtin_amdgcn_s_wait_tensorcnt(0);
    }
#else
    // Fallback: cooperative float4 staging, coalesced.
    for (int idx = tid; idx < Cc * (TL / 4); idx += 256) {
        const int c  = idx >> 5;               // / (TL/4)
        const int j4 = idx & (TL / 4 - 1);
        const f4 v = *(const f4*)(hv + hv_base + (size_t)c * Lc + j4 * 4);
        *(f4*)&tile[c][j4 * 4] = v;
    }
#endif
    for (int d = tid; d < Dc; d += 256) u_sh[d] = ws[16 + d];
    __syncthreads();

    // Per-column softmax pool (threads 0..127 each own one l).
    if (tid < TL) {
        const int j = tid;
        float s = 0.f;
        #pragma unroll 8
        for (int c = 0; c < Cc; ++c) s += tile[c][j];
        const float coef = s * (1.f / (float)Cc) * scale_qk;

        float m = -__builtin_inff();
        #pragma unroll 8
        for (int c = 0; c < Cc; ++c) m = fmaxf(m, tile[c][j] * coef);

        float denom = 0.f, pnum = 0.f;
        #pragma unroll 4
        for (int c = 0; c < Cc; ++c) {
            const float h = tile[c][j];
            const float e = __expf(h * coef - m);
            denom += e;
            pnum  += e * h;
        }
        pooled_sh[j] = pnum / denom;
    }
    __syncthreads();

    // psi[b, d, l0+j] = u[d] * pooled[j]; non-temporal float4 stores
    // (psi is written once, never re-read on device -> keep L2 for h_v).
    const size_t out_base = (size_t)b * Dc * Lc + l0;
    for (int idx = tid; idx < Dc * (TL / 4); idx += 256) {
        const int d  = idx >> 5;
        const int j4 = idx & (TL / 4 - 1);
        const f4 p = *(const f4*)&pooled_sh[j4 * 4];
        const float ud = u_sh[d];
        f4 r; r.x = ud * p.x; r.y = ud * p.y; r.z = ud * p.z; r.w = ud * p.w;
        __builtin_nontemporal_store(r, (f4*)(out + out_base + (size_t)d * Lc + j4 * 4));
    }
}

extern "C" void kernel_launch(void* const* d_in, const int* in_sizes, int n_in,
                              void* d_out, int out_size, void* d_ws, size_t ws_size,
                              hipStream_t stream) {
    const float* h_v   = (const float*)d_in[0];
    const float* wq    = (const float*)d_in[1];
    const float* wk    = (const float*)d_in[2];
    const float* wv    = (const float*)d_in[3];
    const float* w_out = (const float*)d_in[4];
    float* out = (float*)d_out;
    float* ws  = (float*)d_ws;

    iap_prep_kernel<<<1, 256, 0, stream>>>(wq, wk, wv, w_out, ws);
    iap_pool_kernel<<<Bc * (Lc / TL), 256, 0, stream>>>(h_v, ws, out);
}